// SAGE_42528766165146
// MI455X (gfx1250) — compile-verified
//
#include <hip/hip_runtime.h>

// ---------------- problem constants (match reference) ----------------
#define NN   100000      // nodes
#define NE   1600000     // edges
#define DIN  32          // input feature dim
#define HD   64          // hidden dim
#define NL   4           // layers
#define NG   128         // graphs
#define NC   10          // classes
#define BNEPS 1e-5f

typedef float v2f __attribute__((ext_vector_type(2)));
typedef float v8f __attribute__((ext_vector_type(8)));

// ---------------- helpers ----------------
__global__ void fill_kernel(float* __restrict__ p, int n, float v) {
    int i = blockIdx.x * blockDim.x + threadIdx.x;
    if (i < n) p[i] = v;
}

// degree per destination node (float) + node count per graph (float)
__global__ void counts_kernel(const int* __restrict__ dst, const int* __restrict__ batch,
                              float* __restrict__ deg, float* __restrict__ gcnt) {
    int i = blockIdx.x * blockDim.x + threadIdx.x;
    if (i < NE) atomicAdd(&deg[dst[i]], 1.0f);
    if (i < NN) atomicAdd(&gcnt[batch[i]], 1.0f);
}

// agg[dst] += h[src]  (one thread per (edge, 4-feature chunk)); mean's divide
// is folded into the GEMM A-fragment load (scale by 1/deg). agg stays resident
// in the 192MB L2, so these f32 atomics resolve on-chip.
__global__ void scatter_kernel(const float* __restrict__ h, const int* __restrict__ src,
                               const int* __restrict__ dst, float* __restrict__ agg) {
    long t = (long)blockIdx.x * blockDim.x + threadIdx.x;
    if (t >= (long)NE * (HD / 4)) return;
    int e = (int)(t >> 4);
    int c = ((int)t & 15) << 2;
    int s = src[e] * HD + c;
    int d = dst[e] * HD + c;
    float4 v = *(const float4*)(h + s);
    atomicAdd(&agg[d + 0], v.x);
    atomicAdd(&agg[d + 1], v.y);
    atomicAdd(&agg[d + 2], v.z);
    atomicAdd(&agg[d + 3], v.w);
}

// ---------------- fused WMMA GEMM ----------------
// Out[row, 0:64] = (SCALE ? A1*1/deg : A1) @ W1 (+ A2 @ W2) + bias
// A1/A2: [nrows, K] row-major; W1/W2: [K, 64] row-major; exact f32 math via
// V_WMMA_F32_16X16X4_F32 (16x4 A, 4x16 B, 16x16 f32 C/D).
//
// Fragment layouts (CDNA5 ISA 7.12.2):
//   A 16x4 f32 : lanes 0-15 -> M=lane, VGPR0=K0,VGPR1=K1 ; lanes 16-31 -> K2,K3
//   B 4x16 f32 : lanes 0-15 -> N=lane, VGPR0=K0,VGPR1=K1 ; lanes 16-31 -> K2,K3
//   C 16x16    : VGPR r, lane L -> M = r + (L>=16 ? 8:0), N = L%16
//
// Weights are staged into LDS in a fragment-ready interleave:
//   wlds[p*SP + col*2 + j] = W[2p + j][col]   (j = 0,1)
// so a B fragment {W[k][col], W[k+1][col]} is one 8B-aligned ds_load_b64
// straight into a consecutive VGPR pair (no marshaling movs). SP = 160 dwords
// (== 32 mod 64): lane half 0 reads pair-row p, half 1 reads p+1, landing on
// disjoint halves of the 64 LDS banks -> conflict-free b64 fragment loads.
//
// Tail handling is branchless: the A row index is clamped to nrows-1. A-row M
// only ever feeds C-row M, and out-of-range C rows are skipped at store time,
// so the duplicated (finite, in-bounds) data never reaches memory. This keeps
// EXEC fully populated through the k-loop.
#define SP 160   // LDS dwords per k-pair row

template <int K, bool DUAL, bool SCALE>
__global__ __launch_bounds__(256) void sage_gemm_kernel(
    const float* __restrict__ A1, const float* __restrict__ deg,
    const float* __restrict__ W1,
    const float* __restrict__ A2, const float* __restrict__ W2,
    const float* __restrict__ bias,
    float* Out, int nrows)
{
    __shared__ float w1s[(K / 2) * SP];
    __shared__ float w2s[DUAL ? (K / 2) * SP : 1];
    __shared__ float bs[HD];
    for (int i = threadIdx.x; i < K * HD; i += 256) {
        int k = i >> 6, col = i & (HD - 1);
        int d = (k >> 1) * SP + col * 2 + (k & 1);
        w1s[d] = W1[i];
        if (DUAL) w2s[d] = W2[i];
    }
    for (int i = threadIdx.x; i < HD; i += 256) bs[i] = bias[i];
    __syncthreads();

    const int wave = threadIdx.x >> 5;
    const int lane = threadIdx.x & 31;
    const int half = lane >> 4;        // 0: K+0/1, 1: K+2/3
    const int l16  = lane & 15;
    const int row0 = blockIdx.x * 128 + wave * 16;   // 8 waves x 16 rows
    int arow = row0 + l16;                           // row this lane feeds into A
    arow = arow < nrows ? arow : nrows - 1;          // branchless tail clamp

    float scl = 1.0f;
    if (SCALE) scl = 1.0f / fmaxf(deg[arow], 1.0f);

    v8f acc[4];
    for (int t = 0; t < 4; ++t) {
        float b = bs[t * 16 + l16];                  // bias is per-column
        acc[t] = (v8f){b, b, b, b, b, b, b, b};
    }

    const int koff = half * 2;
    const float* ap1 = A1 + (long)arow * K + koff;
    const float* ap2 = DUAL ? (A2 + (long)arow * K + koff) : nullptr;

#pragma unroll
    for (int kk = 0; kk < K; kk += 4) {
        float2 t1 = *(const float2*)(ap1 + kk);
        v2f a1; a1.x = t1.x * scl; a1.y = t1.y * scl;
        v2f a2 = {0.0f, 0.0f};
        if (DUAL) {
            float2 t2 = *(const float2*)(ap2 + kk);
            a2.x = t2.x; a2.y = t2.y;
        }
        const int prow = ((kk >> 1) + half) * SP + l16 * 2;   // fragment base
#pragma unroll
        for (int t = 0; t < 4; ++t) {
            v2f b1 = *(const v2f*)&w1s[prow + t * 32];
            acc[t] = __builtin_amdgcn_wmma_f32_16x16x4_f32(
                false, a1, false, b1, (short)0, acc[t], false, false);
            if (DUAL) {
                v2f b2 = *(const v2f*)&w2s[prow + t * 32];
                acc[t] = __builtin_amdgcn_wmma_f32_16x16x4_f32(
                    false, a2, false, b2, (short)0, acc[t], false, false);
            }
        }
    }

    const int srow = row0 + half * 8;
#pragma unroll
    for (int t = 0; t < 4; ++t)
#pragma unroll
        for (int r = 0; r < 8; ++r) {
            int row = srow + r;
            if (row < nrows) Out[(long)row * HD + t * 16 + l16] = acc[t][r];
        }
}

// ---------------- batchnorm stats: stats[0:64]=sum, stats[64:128]=sumsq ----------------
__global__ void bnstats_kernel(const float* __restrict__ out, float* __restrict__ stats) {
    __shared__ float ssum[HD];
    __shared__ float ssq[HD];
    for (int i = threadIdx.x; i < HD; i += blockDim.x) { ssum[i] = 0.0f; ssq[i] = 0.0f; }
    __syncthreads();
    const long total = (long)NN * HD;
    for (long i = (long)blockIdx.x * blockDim.x + threadIdx.x; i < total;
         i += (long)gridDim.x * blockDim.x) {
        float v = out[i];
        int f = (int)(i & (HD - 1));
        atomicAdd(&ssum[f], v);
        atomicAdd(&ssq[f], v * v);
    }
    __syncthreads();
    for (int i = threadIdx.x; i < HD; i += blockDim.x) {
        atomicAdd(&stats[i], ssum[i]);
        atomicAdd(&stats[HD + i], ssq[i]);
    }
}

// normalize + ReLU + residual (h <- h + relu(bn(out))) + pooled-sum accumulate
__global__ void bnnorm_pool_kernel(const float* __restrict__ out, float* __restrict__ h,
                                   const float* __restrict__ stats,
                                   const float* __restrict__ gamma, const float* __restrict__ beta,
                                   const int* __restrict__ batch, float* __restrict__ featsAcc) {
    long i = (long)blockIdx.x * blockDim.x + threadIdx.x;
    if (i >= (long)NN * HD) return;
    int f = (int)(i & (HD - 1));
    int n = (int)(i >> 6);
    const float inv_n = 1.0f / (float)NN;
    float mu  = stats[f] * inv_n;
    float var = stats[HD + f] * inv_n - mu * mu;
    float o = (out[i] - mu) * rsqrtf(var + BNEPS) * gamma[f] + beta[f];
    o = fmaxf(o, 0.0f);
    float hn = h[i] + o;
    h[i] = hn;
    atomicAdd(&featsAcc[batch[n] * HD + f], hn);
}

__global__ void feats_div_kernel(float* __restrict__ feats, const float* __restrict__ gcnt) {
    int i = blockIdx.x * blockDim.x + threadIdx.x;
    if (i >= NL * NG * HD) return;
    int g = (i >> 6) & (NG - 1);
    feats[i] /= fmaxf(gcnt[g], 1.0f);
}

// tiny per-(layer,graph) MLP head: 64->32 relu ->16 relu ->10
__global__ void heads_kernel(const float* __restrict__ feats,
                             const float* __restrict__ fc0W, const float* __restrict__ fc0b,
                             const float* __restrict__ fc1W, const float* __restrict__ fc1b,
                             const float* __restrict__ fc2W, const float* __restrict__ fc2b,
                             float* __restrict__ outs) {
    int t = blockIdx.x * blockDim.x + threadIdx.x;
    if (t >= NL * NG) return;
    int l = t / NG;
    const float* f = feats + (long)t * HD;
    float f0[32];
    const float* W0 = fc0W + l * HD * 32; const float* b0 = fc0b + l * 32;
    for (int j = 0; j < 32; ++j) {
        float s = b0[j];
        for (int i = 0; i < HD; ++i) s += f[i] * W0[i * 32 + j];
        f0[j] = fmaxf(s, 0.0f);
    }
    float f1[16];
    const float* W1 = fc1W + l * 32 * 16; const float* b1 = fc1b + l * 16;
    for (int j = 0; j < 16; ++j) {
        float s = b1[j];
        for (int i = 0; i < 32; ++i) s += f0[i] * W1[i * 16 + j];
        f1[j] = fmaxf(s, 0.0f);
    }
    const float* W2 = fc2W + l * 16 * NC; const float* b2 = fc2b + l * NC;
    float* op = outs + (long)t * NC;
    for (int j = 0; j < NC; ++j) {
        float s = b2[j];
        for (int i = 0; i < 16; ++i) s += f1[i] * W2[i * NC + j];
        op[j] = s;
    }
}

// ---------------- driver ----------------
extern "C" void kernel_launch(void* const* d_in, const int* in_sizes, int n_in,
                              void* d_out, int out_size, void* d_ws, size_t ws_size,
                              hipStream_t stream) {
    (void)in_sizes; (void)n_in; (void)out_size; (void)ws_size;
    const float* x     = (const float*)d_in[0];
    const int*   edge  = (const int*)d_in[1];
    const int*   src   = edge;
    const int*   dst   = edge + NE;
    const int*   batch = (const int*)d_in[2];
    const float* embW  = (const float*)d_in[3];
    const float* embB  = (const float*)d_in[4];
    const float* Wl    = (const float*)d_in[5];
    const float* Wr    = (const float*)d_in[6];
    const float* convB = (const float*)d_in[7];
    const float* gamma = (const float*)d_in[8];
    const float* beta  = (const float*)d_in[9];
    const float* fc0W  = (const float*)d_in[10];
    const float* fc0b  = (const float*)d_in[11];
    const float* fc1W  = (const float*)d_in[12];
    const float* fc1b  = (const float*)d_in[13];
    const float* fc2W  = (const float*)d_in[14];
    const float* fc2b  = (const float*)d_in[15];

    float* outsRegion  = (float*)d_out;                 // [L,G,C]
    float* featsRegion = outsRegion + NL * NG * NC;     // [L,G,H]

    // workspace layout (floats): h[N*H] | agg/out[N*H] | deg[N] | gcnt[G] | stats[2H]
    float* ws    = (float*)d_ws;
    float* h     = ws;
    float* agg   = h + (long)NN * HD;     // also serves as the GEMM output (safe alias)
    float* deg   = agg + (long)NN * HD;
    float* gcnt  = deg + NN;
    float* stats = gcnt + NG;

    const int T = 256;
    const int gemmGrid = (NN + 127) / 128;

    // one-time zeroing + counts
    fill_kernel<<<(NN + NG + T - 1) / T, T, 0, stream>>>(deg, NN + NG, 0.0f);
    fill_kernel<<<(NL * NG * HD + T - 1) / T, T, 0, stream>>>(featsRegion, NL * NG * HD, 0.0f);
    counts_kernel<<<(NE + T - 1) / T, T, 0, stream>>>(dst, batch, deg, gcnt);

    // embedding: h = x @ embW + embB   (K=32, single matrix)
    sage_gemm_kernel<DIN, false, false><<<gemmGrid, 256, 0, stream>>>(
        x, nullptr, embW, nullptr, nullptr, embB, h, NN);

    for (int l = 0; l < NL; ++l) {
        fill_kernel<<<((int)((long)NN * HD) + T - 1) / T, T, 0, stream>>>(agg, NN * HD, 0.0f);
        fill_kernel<<<1, T, 0, stream>>>(stats, 2 * HD, 0.0f);
        long sthreads = (long)NE * (HD / 4);
        scatter_kernel<<<(int)((sthreads + T - 1) / T), T, 0, stream>>>(h, src, dst, agg);
        // out = mean(agg) @ Wl + h @ Wr + b   (out aliases agg; each wave reads
        // all of its rows before storing them)
        sage_gemm_kernel<HD, true, true><<<gemmGrid, 256, 0, stream>>>(
            agg, deg, Wl + (long)l * HD * HD, h, Wr + (long)l * HD * HD,
            convB + l * HD, agg, NN);
        bnstats_kernel<<<512, 256, 0, stream>>>(agg, stats);
        bnnorm_pool_kernel<<<(int)(((long)NN * HD + T - 1) / T), T, 0, stream>>>(
            agg, h, stats, gamma + l * HD, beta + l * HD, batch,
            featsRegion + (long)l * NG * HD);
    }

    feats_div_kernel<<<(NL * NG * HD + T - 1) / T, T, 0, stream>>>(featsRegion, gcnt);
    heads_kernel<<<(NL * NG + 127) / 128, 128, 0, stream>>>(
        featsRegion, fc0W, fc0b, fc1W, fc1b, fc2W, fc2b, outsRegion);
}